// GraphConv_8040178778539
// MI455X (gfx1250) — compile-verified
//
#include <hip/hip_runtime.h>
#include <hip/hip_bf16.h>
#include <limits.h>

#define N_USERS    50000
#define N_ENTITIES 50000
#define N_NODES    100000
#define CHANNEL    128
#define N_EDGES    1000000
#define N_EXTRA    1000000
#define NNZ_       1000000

typedef float v2f __attribute__((ext_vector_type(2)));
typedef float v8f __attribute__((ext_vector_type(8)));

#define CDIV(a, b) (((a) + (b) - 1) / (b))

static __device__ __forceinline__ float atomAddF(float* p, float v) {
  return __hip_atomic_fetch_add(p, v, __ATOMIC_RELAXED, __HIP_MEMORY_SCOPE_AGENT);
}

// ---------------- fills ----------------
__global__ void k_fill_f32(float* __restrict__ p, float v, int n) {
  int i = blockIdx.x * blockDim.x + threadIdx.x;
  if (i < n) p[i] = v;
}
__global__ void k_fill_i32(int* __restrict__ p, int v, int n) {
  int i = blockIdx.x * blockDim.x + threadIdx.x;
  if (i < n) p[i] = v;
}

// ---------------- node concat + residual init ----------------
// node_a = [user_emb; entity_emb]; d_out[user_res|entity_res] = node_a; d_out[node_res] = node_a
__global__ void k_concat_init(const float* __restrict__ user, const float* __restrict__ ent,
                              float* __restrict__ node_a, float* __restrict__ out) {
  int i = blockIdx.x * blockDim.x + threadIdx.x;
  const int total = N_NODES * CHANNEL;
  if (i >= total) return;
  float v = (i < N_USERS * CHANNEL) ? user[i] : ent[i - N_USERS * CHANNEL];
  node_a[i] = v;
  out[i] = v;                       // gcn_res init = [user;entity]
  out[(size_t)total + i] = v;       // node_res init = node_emb
}

// ---------------- WMMA f32 GEMM: Q = A*WQ, K = A*WK ----------------
// One block = one 16-row M tile. 8 waves, each owns one 16-col N tile, accumulates Q and K.
__global__ __launch_bounds__(256) void k_gemm_qk(const float* __restrict__ A,
                                                 const float* __restrict__ WQ,
                                                 const float* __restrict__ WK,
                                                 float* __restrict__ Q,
                                                 float* __restrict__ K) {
  __shared__ __align__(16) float As[16 * 132];   // stride 132 -> conflict-free ds reads
  const int m0 = blockIdx.x * 16;
  const int tid = threadIdx.x;
  for (int t = 0; t < 8; ++t) {
    int idx = tid + t * 256;
    int r = idx >> 7, c = idx & 127;
    As[r * 132 + c] = A[(size_t)(m0 + r) * CHANNEL + c];
  }
  __syncthreads();
  const int wave = tid >> 5;
  const int lane = tid & 31;
  const int n0 = wave * 16;
  const int lm = lane & 15;   // A: row M ; B/D: col N
  const int lh = lane >> 4;   // K-half selector
  v8f cQ = {};
  v8f cK = {};
  for (int k = 0; k < CHANNEL; k += 4) {
    int ka = k + 2 * lh;
    v2f a = *(const v2f*)(&As[lm * 132 + ka]);
    v2f bq, bk;
    bq.x = WQ[(size_t)ka * CHANNEL + n0 + lm];
    bq.y = WQ[(size_t)(ka + 1) * CHANNEL + n0 + lm];
    bk.x = WK[(size_t)ka * CHANNEL + n0 + lm];
    bk.y = WK[(size_t)(ka + 1) * CHANNEL + n0 + lm];
    cQ = __builtin_amdgcn_wmma_f32_16x16x4_f32(false, a, false, bq, (short)0, cQ, false, false);
    cK = __builtin_amdgcn_wmma_f32_16x16x4_f32(false, a, false, bk, (short)0, cK, false, false);
  }
  for (int r = 0; r < 8; ++r) {
    int m = m0 + r + 8 * lh;    // D layout: VGPR r, lane -> N=lane%16, M=r+8*(lane/16)
    Q[(size_t)m * CHANNEL + n0 + lm] = cQ[r];
    K[(size_t)m * CHANNEL + n0 + lm] = cK[r];
  }
}

// ---------------- attention score per edge (wave per edge) ----------------
__global__ __launch_bounds__(256) void k_score(const float* __restrict__ Q,
                                               const float* __restrict__ K,
                                               const int* __restrict__ head,
                                               const int* __restrict__ tail,
                                               float* __restrict__ score, int E) {
  int w = (blockIdx.x * blockDim.x + threadIdx.x) >> 5;
  int lane = threadIdx.x & 31;
  if (w >= E) return;
  int h = head[w], t = tail[w];
  float4 q = *(const float4*)(Q + (size_t)h * CHANNEL + lane * 4);
  float4 k = *(const float4*)(K + (size_t)t * CHANNEL + lane * 4);
  float p = q.x * k.x + q.y * k.y + q.z * k.z + q.w * k.w;
  p += __shfl_xor(p, 1, 32);
  p += __shfl_xor(p, 2, 32);
  p += __shfl_xor(p, 4, 32);
  p += __shfl_xor(p, 8, 32);       // lanes 0..15 = head0 sum, 16..31 = head1 sum
  if ((lane & 15) == 0) score[(size_t)w * 2 + (lane >> 4)] = p * 0.125f;  // 1/sqrt(64)
}

// monotone float->int key so signed-int atomicMax orders like float
static __device__ __forceinline__ int fkey(float f) {
  int b = __float_as_int(f);
  return (b >= 0) ? b : (b ^ 0x7fffffff);
}
static __device__ __forceinline__ float funkey(int k) {
  return __int_as_float((k >= 0) ? k : (k ^ 0x7fffffff));
}

__global__ void k_segmax(const float* __restrict__ score, const int* __restrict__ head,
                         int* __restrict__ mkey, int n) {   // n = E*2
  int i = blockIdx.x * blockDim.x + threadIdx.x;
  if (i >= n) return;
  int e = i >> 1, h = i & 1;
  atomicMax(&mkey[head[e] * 2 + h], fkey(score[i]));
}

__global__ void k_expz(float* __restrict__ score, const int* __restrict__ head,
                       const int* __restrict__ mkey, float* __restrict__ z, int n) {
  int i = blockIdx.x * blockDim.x + threadIdx.x;
  if (i >= n) return;
  int e = i >> 1, h = i & 1;
  int seg = head[e] * 2 + h;
  float m = funkey(mkey[seg]);
  float ev = expf(score[i] - m);
  score[i] = ev;                   // overwrite scores with exp values
  atomAddF(&z[seg], ev);
}

__global__ void k_augw(const float* __restrict__ escore, const float* __restrict__ z,
                       const int* __restrict__ head, float* __restrict__ aug, int E) {
  int e = blockIdx.x * blockDim.x + threadIdx.x;
  if (e >= E) return;
  int h = head[e];
  float a0 = escore[(size_t)e * 2 + 0] / z[h * 2 + 0];
  float a1 = escore[(size_t)e * 2 + 1] / z[h * 2 + 1];
  aug[e] = 0.5f * (a0 + a1);
}

// ---------------- segment counts (fixed across hops) ----------------
__global__ void k_count(const int* __restrict__ head, float* __restrict__ cnt, int E) {
  int e = blockIdx.x * blockDim.x + threadIdx.x;
  if (e >= E) return;
  atomAddF(&cnt[head[e]], 1.0f);
}

// ---------------- KG aggregation: ent_out[head] += ent_in[tail]*weight[type-1] ----------------
__global__ __launch_bounds__(256) void k_kg_agg(const float* __restrict__ ent_in,
                                                const int* __restrict__ head,
                                                const int* __restrict__ tail,
                                                const int* __restrict__ etype,
                                                const float* __restrict__ weight,
                                                float* __restrict__ ent_out, int E) {
  int w = (blockIdx.x * blockDim.x + threadIdx.x) >> 5;
  int lane = threadIdx.x & 31;
  if (w >= E) return;
  int hd = head[w], tl = tail[w], r = etype[w] - 1;
  float4 x = *(const float4*)(ent_in + (size_t)tl * CHANNEL + lane * 4);
  float4 ww = *(const float4*)(weight + (size_t)r * CHANNEL + lane * 4);
  float* dst = ent_out + (size_t)hd * CHANNEL + lane * 4;
  atomAddF(dst + 0, x.x * ww.x);
  atomAddF(dst + 1, x.y * ww.y);
  atomAddF(dst + 2, x.z * ww.z);
  atomAddF(dst + 3, x.w * ww.w);
}

// ---------------- preference aggregation: node_out[head] += node_in[tail]*aug_w[e] ----------------
__global__ __launch_bounds__(256) void k_pref_agg(const float* __restrict__ node_in,
                                                  const int* __restrict__ head,
                                                  const int* __restrict__ tail,
                                                  const float* __restrict__ aug,
                                                  float* __restrict__ node_out, int E) {
  int w = (blockIdx.x * blockDim.x + threadIdx.x) >> 5;
  int lane = threadIdx.x & 31;
  if (w >= E) return;
  int hd = head[w], tl = tail[w];
  float a = aug[w];
  float4 x = *(const float4*)(node_in + (size_t)tl * CHANNEL + lane * 4);
  float* dst = node_out + (size_t)hd * CHANNEL + lane * 4;
  atomAddF(dst + 0, x.x * a);
  atomAddF(dst + 1, x.y * a);
  atomAddF(dst + 2, x.z * a);
  atomAddF(dst + 3, x.w * a);
}

// ---------------- interact SpMM: user[row] += val * ent[col] ----------------
__global__ __launch_bounds__(256) void k_interact(const float* __restrict__ ent,
                                                  const int* __restrict__ rows,
                                                  const int* __restrict__ cols,
                                                  const float* __restrict__ vals,
                                                  float* __restrict__ user, int NNZv) {
  int w = (blockIdx.x * blockDim.x + threadIdx.x) >> 5;
  int lane = threadIdx.x & 31;
  if (w >= NNZv) return;
  int rr = rows[w], cc = cols[w];
  float v = vals[w];
  float4 x = *(const float4*)(ent + (size_t)cc * CHANNEL + lane * 4);
  float* dst = user + (size_t)rr * CHANNEL + lane * 4;
  atomAddF(dst + 0, x.x * v);
  atomAddF(dst + 1, x.y * v);
  atomAddF(dst + 2, x.z * v);
  atomAddF(dst + 3, x.w * v);
}

// ---------------- mean divide ----------------
__global__ void k_div(float* __restrict__ buf, const float* __restrict__ cnt, int n) {
  int i = blockIdx.x * blockDim.x + threadIdx.x;
  if (i >= n) return;
  buf[i] = buf[i] / fmaxf(cnt[i >> 7], 1.0f);
}

// ---------------- l2 normalize (in place) + residual accumulate (wave per row) ----------------
__global__ __launch_bounds__(256) void k_l2norm_acc(float* __restrict__ buf,
                                                    float* __restrict__ res, int rows) {
  int w = (blockIdx.x * blockDim.x + threadIdx.x) >> 5;
  int lane = threadIdx.x & 31;
  if (w >= rows) return;
  float* p = buf + (size_t)w * CHANNEL + lane * 4;
  float4 x = *(const float4*)p;
  float ss = x.x * x.x + x.y * x.y + x.z * x.z + x.w * x.w;
  ss += __shfl_xor(ss, 1, 32);
  ss += __shfl_xor(ss, 2, 32);
  ss += __shfl_xor(ss, 4, 32);
  ss += __shfl_xor(ss, 8, 32);
  ss += __shfl_xor(ss, 16, 32);
  float inv = 1.0f / fmaxf(sqrtf(ss), 1e-12f);
  x.x *= inv; x.y *= inv; x.z *= inv; x.w *= inv;
  *(float4*)p = x;
  float* r = res + (size_t)w * CHANNEL + lane * 4;
  r[0] += x.x; r[1] += x.y; r[2] += x.z; r[3] += x.w;
}

// =====================================================================
extern "C" void kernel_launch(void* const* d_in, const int* in_sizes, int n_in,
                              void* d_out, int out_size, void* d_ws, size_t ws_size,
                              hipStream_t stream) {
  const float* user_emb  = (const float*)d_in[0];
  const float* ent_emb   = (const float*)d_in[1];
  const int*   ent_head  = (const int*)d_in[2];            // edge_index[0]
  const int*   ent_tail  = ((const int*)d_in[2]) + N_EDGES; // edge_index[1]
  const int*   ent_type  = (const int*)d_in[3];
  const int*   ex_head   = (const int*)d_in[4];
  const int*   ex_tail   = ((const int*)d_in[4]) + N_EXTRA;
  const int*   irows     = (const int*)d_in[6];
  const int*   icols     = (const int*)d_in[7];
  const float* ivals     = (const float*)d_in[8];
  const float* weight    = (const float*)d_in[9];
  const float* W_Q       = (const float*)d_in[11];
  const float* W_K       = (const float*)d_in[12];
  float* out = (float*)d_out;

  // workspace layout (floats)
  float* ws = (float*)d_ws;
  const size_t NC = (size_t)N_NODES * CHANNEL;     // 12.8M
  const size_t EC = (size_t)N_ENTITIES * CHANNEL;  // 6.4M
  float* node_a  = ws;                  // initial node emb / hop2 node_out
  float* node_b  = ws + NC;             // Qn, then hop1 node_out
  float* kbuf    = ws + 2 * NC;         // Kn, then ent_a | ent_b
  float* ent_a   = kbuf;
  float* ent_b   = kbuf + EC;
  float* userb   = ws + 3 * NC;                     // 6.4M
  float* score   = userb + EC;                      // 2M (E*2)
  float* aug     = score + 2 * (size_t)N_EXTRA;     // 1M
  float* z       = aug + N_EXTRA;                   // 200K
  int*   mkey    = (int*)(z + 2 * (size_t)N_NODES); // 200K
  float* cnt_nd  = (float*)(mkey + 2 * (size_t)N_NODES); // 100K
  float* cnt_en  = cnt_nd + N_NODES;                     // 50K

  const int B = 256;

  // --- concat + residual init ---
  k_concat_init<<<CDIV((int)NC, B), B, 0, stream>>>(user_emb, ent_emb, node_a, out);

  // --- Q/K projection (WMMA) ---
  k_gemm_qk<<<N_NODES / 16, 256, 0, stream>>>(node_a, W_Q, W_K, node_b, kbuf);

  // --- attention softmax over head segments ---
  k_fill_i32<<<CDIV(2 * N_NODES, B), B, 0, stream>>>(mkey, INT_MIN, 2 * N_NODES);
  k_fill_f32<<<CDIV(2 * N_NODES, B), B, 0, stream>>>(z, 0.0f, 2 * N_NODES);
  k_score<<<CDIV(N_EXTRA * 32, B), B, 0, stream>>>(node_b, kbuf, ex_head, ex_tail, score, N_EXTRA);
  k_segmax<<<CDIV(2 * N_EXTRA, B), B, 0, stream>>>(score, ex_head, mkey, 2 * N_EXTRA);
  k_expz<<<CDIV(2 * N_EXTRA, B), B, 0, stream>>>(score, ex_head, mkey, z, 2 * N_EXTRA);
  k_augw<<<CDIV(N_EXTRA, B), B, 0, stream>>>(score, z, ex_head, aug, N_EXTRA);

  // --- segment counts (hop-invariant) ---
  k_fill_f32<<<CDIV(N_NODES, B), B, 0, stream>>>(cnt_nd, 0.0f, N_NODES);
  k_fill_f32<<<CDIV(N_ENTITIES, B), B, 0, stream>>>(cnt_en, 0.0f, N_ENTITIES);
  k_count<<<CDIV(N_EXTRA, B), B, 0, stream>>>(ex_head, cnt_nd, N_EXTRA);
  k_count<<<CDIV(N_EDGES, B), B, 0, stream>>>(ent_head, cnt_en, N_EDGES);

  // --- two hops, ping-pong buffers ---
  const float* ent_in[2]  = { ent_emb, ent_a };
  float*       ent_out[2] = { ent_a,   ent_b };
  const float* nd_in[2]   = { node_a,  node_b };
  float*       nd_out[2]  = { node_b,  node_a };

  for (int hop = 0; hop < 2; ++hop) {
    k_fill_f32<<<CDIV((int)EC, B), B, 0, stream>>>(ent_out[hop], 0.0f, (int)EC);
    k_fill_f32<<<CDIV((int)NC, B), B, 0, stream>>>(nd_out[hop], 0.0f, (int)NC);
    k_fill_f32<<<CDIV((int)EC, B), B, 0, stream>>>(userb, 0.0f, (int)EC);

    k_kg_agg<<<CDIV(N_EDGES * 32, B), B, 0, stream>>>(ent_in[hop], ent_head, ent_tail,
                                                      ent_type, weight, ent_out[hop], N_EDGES);
    k_div<<<CDIV((int)EC, B), B, 0, stream>>>(ent_out[hop], cnt_en, (int)EC);

    k_pref_agg<<<CDIV(N_EXTRA * 32, B), B, 0, stream>>>(nd_in[hop], ex_head, ex_tail,
                                                        aug, nd_out[hop], N_EXTRA);
    k_div<<<CDIV((int)NC, B), B, 0, stream>>>(nd_out[hop], cnt_nd, (int)NC);

    // user = interact_mat @ entity (post-mean, pre-norm)
    k_interact<<<CDIV(NNZ_ * 32, B), B, 0, stream>>>(ent_out[hop], irows, icols, ivals,
                                                     userb, NNZ_);

    // normalize in place + accumulate residuals into d_out regions
    k_l2norm_acc<<<CDIV(N_ENTITIES * 32, B), B, 0, stream>>>(ent_out[hop], out + EC, N_ENTITIES);
    k_l2norm_acc<<<CDIV(N_NODES * 32, B), B, 0, stream>>>(nd_out[hop], out + NC, N_NODES);
    k_l2norm_acc<<<CDIV(N_USERS * 32, B), B, 0, stream>>>(userb, out, N_USERS);
  }
}